// GraphSAGELayer_21294447853583
// MI455X (gfx1250) — compile-verified
//
#include <hip/hip_runtime.h>

// ---------------------------------------------------------------------------
// GraphSAGE layer, fused:  out = relu([x, adj@x] @ W^T + b)
//   x   [16384, 128] f32, adj [16384, 16384] f32,
//   W   [128, 256]  f32,  b  [128] f32,  out [16384, 128] f32
// One workgroup = 128-row strip; h_N strip kept on-chip; WMMA f32 16x16x4.
// B-side LDS operands use a K-pair interleaved layout so each WMMA B fragment
// is a single aligned b64 LDS load (no v_mov packing).
// ---------------------------------------------------------------------------

typedef float v2f __attribute__((ext_vector_type(2)));
typedef float v8f __attribute__((ext_vector_type(8)));

#define NROW 16384
#define DIM  128
#define KT   64      // K tile of the adj@x loop
#define BM   128     // rows per workgroup
#define AS   68      // LDS stride: adj tile rows (A frags conflict-free)
#define HS   132     // LDS stride: hN / x strip  (A frags in epilogue)
#define XSP  288     // paired stride: x tile, 2*DIM+32  (288%64==32)
#define WSP  288     // paired stride: W^T tile

// LDS float layout:
//   region0 (union):
//     main loop : A_s (128 x 68 = 8704)  then  Xp (32 pair-rows x 288 = 9216)
//     epilogue  : Wt  (64 pair-rows x 288 = 18432)
//   region1     : H_s (128 x 132 = 16896)   hN strip, later x rows
#define R0_FLOATS 18432
#define SMEM_FLOATS (R0_FLOATS + BM * HS)

__device__ __forceinline__ v8f wmma_f32(v2f a, v2f b, v8f c) {
  // (neg_a, A, neg_b, B, c_mod, C, reuse_a, reuse_b)
  return __builtin_amdgcn_wmma_f32_16x16x4_f32(false, a, false, b, (short)0, c,
                                               false, false);
}

__global__ __launch_bounds__(256)
void sage_fused_wmma(const float* __restrict__ x, const float* __restrict__ adj,
                     const float* __restrict__ W, const float* __restrict__ bias,
                     float* __restrict__ out) {
  __shared__ float smem[SMEM_FLOATS];
  float* A_s = smem;                   // 128 x AS
  float* Xp  = smem + BM * AS;         // 32 pair-rows x XSP
  float* Wt  = smem;                   // epilogue alias: 64 pair-rows x WSP
  float* H_s = smem + R0_FLOATS;       // 128 x HS

  const int tid  = threadIdx.x;
  const int lane = tid & 31;
  const int wave = tid >> 5;
  const int half = lane >> 4;   // which 16-lane half of the wave
  const int m    = lane & 15;   // M (A rows) / N (B cols) index inside a tile
  const int wr   = (wave & 3) * 32;   // wave row offset  (4 groups x 32 rows)
  const int wc   = (wave >> 2) * 64;  // wave col offset  (2 groups x 64 cols)
  const int rowBase = blockIdx.x * BM;

  v8f acc[2][4];
#pragma unroll
  for (int i = 0; i < 2; ++i)
#pragma unroll
    for (int j = 0; j < 4; ++j) acc[i][j] = (v8f){0, 0, 0, 0, 0, 0, 0, 0};

  // ---------------- main loop: h_N strip = adj[rows,:] @ x ----------------
  for (int k0 = 0; k0 < NROW; k0 += KT) {
    // stage adj tile: 128 rows x 64 k  (8 float4 per thread, coalesced)
#pragma unroll
    for (int i = 0; i < 8; ++i) {
      const int idx = tid + i * 256;
      const int c4 = idx & 15, r = idx >> 4;
      const float4 v =
          *(const float4*)(adj + (size_t)(rowBase + r) * NROW + k0 + c4 * 4);
      *(float4*)(A_s + r * AS + c4 * 4) = v;
    }
    // stage x tile in K-pair interleaved form:
    //   Xp[(k>>1)*XSP + 2*c + (k&1)] = x[k0+k][c]
    // each thread loads two adjacent k-rows and writes {k, k+1} pairs as b64
#pragma unroll
    for (int p = 0; p < 4; ++p) {
      const int idx = tid + p * 256;        // 32 pair-rows x 32 c4 chunks
      const int c4 = idx & 31, rp = idx >> 5;
      const float4 v0 = *(const float4*)(x + (k0 + 2 * rp) * DIM + c4 * 4);
      const float4 v1 = *(const float4*)(x + (k0 + 2 * rp + 1) * DIM + c4 * 4);
      float* dst = Xp + rp * XSP + c4 * 8;
      *(v2f*)(dst + 0) = (v2f){v0.x, v1.x};
      *(v2f*)(dst + 2) = (v2f){v0.y, v1.y};
      *(v2f*)(dst + 4) = (v2f){v0.z, v1.z};
      *(v2f*)(dst + 6) = (v2f){v0.w, v1.w};
    }
    // hint-prefetch the next adj tile (global_prefetch_b8)
    if (k0 + KT < NROW) {
      const int r = tid >> 1, c = (tid & 1) * 32;
      __builtin_prefetch(adj + (size_t)(rowBase + r) * NROW + (k0 + KT) + c, 0, 0);
    }
    __syncthreads();

#pragma unroll 4
    for (int kk = 0; kk < KT; kk += 4) {
      const int ka = kk + half * 2;          // A frag K base for this half
      const int kp = (kk >> 1) + half;       // B pair-row for this half
      v2f a0, a1;
      a0.x = A_s[(wr + m) * AS + ka];
      a0.y = A_s[(wr + m) * AS + ka + 1];
      a1.x = A_s[(wr + 16 + m) * AS + ka];
      a1.y = A_s[(wr + 16 + m) * AS + ka + 1];
#pragma unroll
      for (int j = 0; j < 4; ++j) {
        const v2f bb = *(const v2f*)(Xp + kp * XSP + 2 * (wc + j * 16 + m));
        acc[0][j] = wmma_f32(a0, bb, acc[0][j]);
        acc[1][j] = wmma_f32(a1, bb, acc[1][j]);
      }
    }
    __syncthreads();
  }

  // ---------------- epilogue: out = relu(x@W1^T + hN@W2^T + b) ------------
  // spill hN accumulators into H_s  (C layout: row = v + 8*half)
#pragma unroll
  for (int i = 0; i < 2; ++i)
#pragma unroll
    for (int j = 0; j < 4; ++j)
#pragma unroll
      for (int v = 0; v < 8; ++v)
        H_s[(wr + i * 16 + v + 8 * half) * HS + wc + j * 16 + m] = acc[i][j][v];

  // load W2^T (cols 128..255 of W) transposed + pair-interleaved into Wt:
  //   Wt[(d>>1)*WSP + 2*o + (d&1)] = W[o][128 + d]
#pragma unroll
  for (int i = 0; i < 16; ++i) {
    const int idx = tid + i * 256;  // 128 outs x 32 d4
    const int d4 = idx & 31, o = idx >> 5;
    const float4 v = *(const float4*)(W + o * (2 * DIM) + DIM + d4 * 4);
    *(v2f*)(Wt + (2 * d4 + 0) * WSP + 2 * o) = (v2f){v.x, v.y};
    *(v2f*)(Wt + (2 * d4 + 1) * WSP + 2 * o) = (v2f){v.z, v.w};
  }
  __syncthreads();

  v8f acc2[2][4];
#pragma unroll
  for (int i = 0; i < 2; ++i)
#pragma unroll
    for (int j = 0; j < 4; ++j) acc2[i][j] = (v8f){0, 0, 0, 0, 0, 0, 0, 0};

  // pass 1: acc2 += hN @ W2^T   (K = 128, from H_s / Wt)
#pragma unroll 2
  for (int kk = 0; kk < DIM; kk += 4) {
    const int ka = kk + half * 2;
    const int kp = (kk >> 1) + half;
    v2f a0, a1;
    a0.x = H_s[(wr + m) * HS + ka];
    a0.y = H_s[(wr + m) * HS + ka + 1];
    a1.x = H_s[(wr + 16 + m) * HS + ka];
    a1.y = H_s[(wr + 16 + m) * HS + ka + 1];
#pragma unroll
    for (int j = 0; j < 4; ++j) {
      const v2f bb = *(const v2f*)(Wt + kp * WSP + 2 * (wc + j * 16 + m));
      acc2[0][j] = wmma_f32(a0, bb, acc2[0][j]);
      acc2[1][j] = wmma_f32(a1, bb, acc2[1][j]);
    }
  }
  __syncthreads();

  // overwrite H_s with this strip's x rows, Wt with W1^T (cols 0..127)
#pragma unroll
  for (int i = 0; i < 16; ++i) {
    const int idx = tid + i * 256;  // 128 rows x 32 c4
    const int c4 = idx & 31, r = idx >> 5;
    const float4 v = *(const float4*)(x + (rowBase + r) * DIM + c4 * 4);
    *(float4*)(H_s + r * HS + c4 * 4) = v;
  }
#pragma unroll
  for (int i = 0; i < 16; ++i) {
    const int idx = tid + i * 256;
    const int d4 = idx & 31, o = idx >> 5;
    const float4 v = *(const float4*)(W + o * (2 * DIM) + d4 * 4);
    *(v2f*)(Wt + (2 * d4 + 0) * WSP + 2 * o) = (v2f){v.x, v.y};
    *(v2f*)(Wt + (2 * d4 + 1) * WSP + 2 * o) = (v2f){v.z, v.w};
  }
  __syncthreads();

  // pass 2: acc2 += x @ W1^T
#pragma unroll 2
  for (int kk = 0; kk < DIM; kk += 4) {
    const int ka = kk + half * 2;
    const int kp = (kk >> 1) + half;
    v2f a0, a1;
    a0.x = H_s[(wr + m) * HS + ka];
    a0.y = H_s[(wr + m) * HS + ka + 1];
    a1.x = H_s[(wr + 16 + m) * HS + ka];
    a1.y = H_s[(wr + 16 + m) * HS + ka + 1];
#pragma unroll
    for (int j = 0; j < 4; ++j) {
      const v2f bb = *(const v2f*)(Wt + kp * WSP + 2 * (wc + j * 16 + m));
      acc2[0][j] = wmma_f32(a0, bb, acc2[0][j]);
      acc2[1][j] = wmma_f32(a1, bb, acc2[1][j]);
    }
  }

  // bias + relu + store
#pragma unroll
  for (int j = 0; j < 4; ++j) {
    const int col = wc + j * 16 + m;
    const float bv = bias[col];
#pragma unroll
    for (int i = 0; i < 2; ++i) {
#pragma unroll
      for (int v = 0; v < 8; ++v) {
        const int row = rowBase + wr + i * 16 + v + 8 * half;
        const float val = acc2[i][j][v] + bv;
        out[row * DIM + col] = fmaxf(val, 0.0f);
      }
    }
  }
}

extern "C" void kernel_launch(void* const* d_in, const int* in_sizes, int n_in,
                              void* d_out, int out_size, void* d_ws,
                              size_t ws_size, hipStream_t stream) {
  (void)in_sizes; (void)n_in; (void)out_size; (void)d_ws; (void)ws_size;
  const float* x   = (const float*)d_in[0];
  const float* adj = (const float*)d_in[1];
  const float* W   = (const float*)d_in[2];
  const float* b   = (const float*)d_in[3];
  float* out = (float*)d_out;

  dim3 grid(NROW / BM);  // 128 workgroups, each a 128-row strip
  dim3 block(256);       // 8 wave32s
  sage_fused_wmma<<<grid, block, 0, stream>>>(x, adj, W, b, out);
}